// NetVLAD_1554778161242
// MI455X (gfx1250) — compile-verified
//
#include <hip/hip_runtime.h>
#include <math.h>

// ---- problem constants ----
#define NB   32
#define CCH  512
#define KCL  64
#define HWP  1200           // 30*40
#define DOUT (KCL*CCH)      // 32768
#define EPSN 1e-12f
#define HTW  48             // hw tile width for logits kernel (1200 = 25*48)
#define NT2  25             // hw tiles for logits kernel

typedef __attribute__((ext_vector_type(2))) float v2f;
typedef __attribute__((ext_vector_type(8))) float v8f;

// ---- CDNA5 async global->LDS copy (ASYNCcnt-tracked) ------------------------
// Generic pointer to LDS: addr[31:0] is the wave-relative LDS byte offset
// (ISA 10.2 aperture rules), which is exactly what VDST of
// GLOBAL_LOAD_ASYNC_TO_LDS_B32 expects.
__device__ __forceinline__ unsigned lds_off(const void* p) {
    return (unsigned)(uintptr_t)p;
}
__device__ __forceinline__ void async_g2l_b32(unsigned lds, const void* gaddr) {
    asm volatile("global_load_async_to_lds_b32 %0, %1, off"
                 :: "v"(lds), "v"(gaddr) : "memory");
}
#define ASYNC_WAIT(N) asm volatile("s_wait_asynccnt %0" :: "n"(N) : "memory")

// -------------------------------------------------------------------------
// Kernel 1: per-pixel inverse L2 norm over channels.  rnorm[n*HW+hw]
// -------------------------------------------------------------------------
__global__ void k_rnorm(const float* __restrict__ x, float* __restrict__ rnorm) {
    int n  = blockIdx.y;
    int hw = blockIdx.x * blockDim.x + threadIdx.x;
    if (hw >= HWP) return;
    const float* p = x + (size_t)n * CCH * HWP + hw;
    float s = 0.f;
#pragma unroll 8
    for (int c = 0; c < CCH; ++c) {
        float v = p[(size_t)c * HWP];
        s += v * v;
    }
    rnorm[n * HWP + hw] = 1.f / fmaxf(sqrtf(s), EPSN);
}

// -------------------------------------------------------------------------
// Kernel 2: logits = conv_w(64x512) @ xn(512x1200), softmax over K.
// Block = 128 thr (4 waves), tile M=64 x N=48 hw; wave w owns k-rows [16w,16w+16)
// and loops 3 hw sub-tiles.  K-loop: C in chunks of 16, double-buffered LDS
// filled with async copies.  B fragments scaled by per-lane-constant rnorm.
// Outputs: a_scaled[n,k,hw] = softmax * rnorm[hw]  (A operand of GEMM-2),
//          ps[n,tile,k]     = per-tile sum of unscaled softmax (for asum).
// -------------------------------------------------------------------------
__global__ void k_logits_softmax(const float* __restrict__ x,
                                 const float* __restrict__ conv_w,
                                 const float* __restrict__ rnorm,
                                 float* __restrict__ a_out,
                                 float* __restrict__ ps) {
    __shared__ float sw[2][64][17];   // conv_w tile [k][c-chunk]
    __shared__ float bx[2][16][HTW + 1]; // raw x tile [c-chunk][hw]
    __shared__ float lg[64][HTW + 1];    // logits tile
    __shared__ float cmax[HTW], crsum[HTW];

    const int n    = blockIdx.y;
    const int hw0  = blockIdx.x * HTW;
    const int t    = threadIdx.x;
    const int wave = t >> 5;
    const int lane = t & 31;
    const int lh   = lane >> 4;
    const int lm   = lane & 15;

    const float* xb = x + (size_t)n * CCH * HWP;
    const float* rn = rnorm + n * HWP + hw0;

    // per-lane-constant column scale factors (N column fixed for whole K loop)
    float rnv0 = rn[lm], rnv1 = rn[16 + lm], rnv2 = rn[32 + lm];

    v8f acc0 = {0.f,0.f,0.f,0.f,0.f,0.f,0.f,0.f};
    v8f acc1 = acc0, acc2 = acc0;

    // async-stage one 16-channel chunk into buffer `buf` (14 ops / thread)
    auto stage = [&](int c0, int buf) {
#pragma unroll
        for (int j = 0; j < 8; ++j) {        // sw: 64x16 = 1024 elems
            int i = t + j * 128, kk = i >> 4, cc = i & 15;
            async_g2l_b32(lds_off(&sw[buf][kk][cc]), conv_w + kk * CCH + c0 + cc);
        }
#pragma unroll
        for (int j = 0; j < 6; ++j) {        // bx: 16x48 = 768 elems
            int i = t + j * 128, cc = i / HTW, hh = i % HTW;
            async_g2l_b32(lds_off(&bx[buf][cc][hh]),
                          xb + (size_t)(c0 + cc) * HWP + hw0 + hh);
        }
    };

    stage(0, 0);
    for (int ch = 0; ch < CCH / 16; ++ch) {
        const int cur = ch & 1;
        if (ch + 1 < CCH / 16) { stage((ch + 1) * 16, cur ^ 1); ASYNC_WAIT(14); }
        else                   { ASYNC_WAIT(0); }
        __syncthreads();
#pragma unroll
        for (int ks = 0; ks < 16; ks += 4) {
            v2f av, b0, b1, b2;
            av[0] = sw[cur][wave * 16 + lm][ks + lh * 2 + 0];
            av[1] = sw[cur][wave * 16 + lm][ks + lh * 2 + 1];
            b0[0] = bx[cur][ks + lh * 2 + 0][lm] * rnv0;
            b0[1] = bx[cur][ks + lh * 2 + 1][lm] * rnv0;
            b1[0] = bx[cur][ks + lh * 2 + 0][16 + lm] * rnv1;
            b1[1] = bx[cur][ks + lh * 2 + 1][16 + lm] * rnv1;
            b2[0] = bx[cur][ks + lh * 2 + 0][32 + lm] * rnv2;
            b2[1] = bx[cur][ks + lh * 2 + 1][32 + lm] * rnv2;
            acc0 = __builtin_amdgcn_wmma_f32_16x16x4_f32(false, av, false, b0,
                                                         (short)0, acc0, false, false);
            acc1 = __builtin_amdgcn_wmma_f32_16x16x4_f32(false, av, false, b1,
                                                         (short)0, acc1, false, false);
            acc2 = __builtin_amdgcn_wmma_f32_16x16x4_f32(false, av, false, b2,
                                                         (short)0, acc2, false, false);
        }
        __syncthreads();
    }

    // spill logits: D layout M = r + 8*lh, N = lm (+ sub-tile offset)
#pragma unroll
    for (int r = 0; r < 8; ++r) {
        lg[wave * 16 + r + 8 * lh][lm]      = acc0[r];
        lg[wave * 16 + r + 8 * lh][16 + lm] = acc1[r];
        lg[wave * 16 + r + 8 * lh][32 + lm] = acc2[r];
    }
    __syncthreads();

    // numerically-stable softmax over 64 clusters, per hw column
    if (t < HTW) {
        float m = -3.402823e38f;
        for (int kk = 0; kk < 64; ++kk) m = fmaxf(m, lg[kk][t]);
        float s = 0.f;
        for (int kk = 0; kk < 64; ++kk) s += expf(lg[kk][t] - m);
        cmax[t]  = m;
        crsum[t] = 1.f / s;
    }
    __syncthreads();

    // per-tile partial sums of UNscaled softmax (for asum)
    if (t < 64) {
        float s = 0.f;
        for (int hh = 0; hh < HTW; ++hh)
            s += expf(lg[t][hh] - cmax[hh]) * crsum[hh];
        ps[((size_t)n * NT2 + blockIdx.x) * KCL + t] = s;
    }
    // coalesced store of rnorm-scaled assignments
    for (int i = t; i < 64 * HTW; i += 128) {
        int kk = i / HTW, hh = i % HTW;
        a_out[((size_t)n * KCL + kk) * HWP + hw0 + hh] =
            expf(lg[kk][hh] - cmax[hh]) * crsum[hh] * rn[hh];
    }
}

// -------------------------------------------------------------------------
// Kernel 3: asum[n,k] = sum over 25 tiles of partial sums (fixed order)
// -------------------------------------------------------------------------
__global__ void k_asum(const float* __restrict__ ps, float* __restrict__ asum) {
    const int n = blockIdx.x, k = threadIdx.x;   // 64 threads
    float s = 0.f;
    for (int j = 0; j < NT2; ++j) s += ps[((size_t)n * NT2 + j) * KCL + k];
    asum[n * KCL + k] = s;
}

// -------------------------------------------------------------------------
// Kernel 4: vlad[n,k,c] = sum_hw a_scaled[k,hw]*x[c,hw] - asum[k]*centroids[k,c]
// Block = 256 thr (8 waves), tile 64k x 64c.  Wave w: k-tile (w>>1), two
// 16-wide c sub-tiles at (w&1)*32 (+16) -> A-fragment reused for 2 WMMAs.
// HW reduction in chunks of 16, double-buffered async LDS staging (raw bytes).
// -------------------------------------------------------------------------
__global__ void k_vlad(const float* __restrict__ x,
                       const float* __restrict__ a_in,  // already * rnorm
                       const float* __restrict__ asum,
                       const float* __restrict__ centroids,
                       float* __restrict__ out) {
    __shared__ float sa[2][64][17];   // a tile [k][hw-chunk]
    __shared__ float bx[2][16][65];   // x^T tile [hw-chunk][c]

    const int n    = blockIdx.y;
    const int c0   = blockIdx.x * 64;
    const int t    = threadIdx.x;
    const int wave = t >> 5;
    const int lane = t & 31;
    const int lh   = lane >> 4;
    const int lm   = lane & 15;
    const int km0  = (wave >> 1) * 16;
    const int cl0  = (wave & 1) * 32;

    const float* xb = x + (size_t)n * CCH * HWP;
    const float* ab = a_in + (size_t)n * KCL * HWP;

    v8f acc0 = {0.f,0.f,0.f,0.f,0.f,0.f,0.f,0.f};
    v8f acc1 = acc0;

    // async-stage one 16-hw chunk (8 ops / thread)
    auto stage = [&](int h0, int buf) {
#pragma unroll
        for (int j = 0; j < 4; ++j) {        // sa: 64x16 = 1024
            int i = t + j * 256, kk = i >> 4, hh = i & 15;
            async_g2l_b32(lds_off(&sa[buf][kk][hh]),
                          ab + (size_t)kk * HWP + h0 + hh);
        }
#pragma unroll
        for (int j = 0; j < 4; ++j) {        // bx: 64c x 16hw = 1024, transposed
            int i = t + j * 256, cc = i >> 4, hh = i & 15;
            async_g2l_b32(lds_off(&bx[buf][hh][cc]),
                          xb + (size_t)(c0 + cc) * HWP + h0 + hh);
        }
    };

    stage(0, 0);
    for (int chk = 0; chk < HWP / 16; ++chk) {
        const int cur = chk & 1;
        if (chk + 1 < HWP / 16) { stage((chk + 1) * 16, cur ^ 1); ASYNC_WAIT(8); }
        else                    { ASYNC_WAIT(0); }
        __syncthreads();
#pragma unroll
        for (int ks = 0; ks < 16; ks += 4) {
            v2f av, b0, b1;
            av[0] = sa[cur][km0 + lm][ks + lh * 2 + 0];
            av[1] = sa[cur][km0 + lm][ks + lh * 2 + 1];
            b0[0] = bx[cur][ks + lh * 2 + 0][cl0 + lm];
            b0[1] = bx[cur][ks + lh * 2 + 1][cl0 + lm];
            b1[0] = bx[cur][ks + lh * 2 + 0][cl0 + 16 + lm];
            b1[1] = bx[cur][ks + lh * 2 + 1][cl0 + 16 + lm];
            acc0 = __builtin_amdgcn_wmma_f32_16x16x4_f32(false, av, false, b0,
                                                         (short)0, acc0, false, false);
            acc1 = __builtin_amdgcn_wmma_f32_16x16x4_f32(false, av, false, b1,
                                                         (short)0, acc1, false, false);
        }
        __syncthreads();
    }

    // epilogue: subtract centroid term, store raw vlad into d_out
#pragma unroll
    for (int r = 0; r < 8; ++r) {
        int kk = km0 + r + 8 * lh;
        float as = asum[n * KCL + kk];
        int cc0 = c0 + cl0 + lm;
        out[((size_t)n * KCL + kk) * CCH + cc0] =
            acc0[r] - as * centroids[kk * CCH + cc0];
        out[((size_t)n * KCL + kk) * CCH + cc0 + 16] =
            acc1[r] - as * centroids[kk * CCH + cc0 + 16];
    }
}

// -------------------------------------------------------------------------
// Kernel 5a: rowss[n,k] = sum_c vlad[n,k,c]^2
// -------------------------------------------------------------------------
__global__ void k_rowss(const float* __restrict__ out, float* __restrict__ rowss) {
    __shared__ float red[128];
    const int nk = blockIdx.x;
    const int t  = threadIdx.x;
    const float* p = out + (size_t)nk * CCH;
    float s = 0.f;
#pragma unroll
    for (int j = 0; j < 4; ++j) { float v = p[t + j * 128]; s += v * v; }
    red[t] = s; __syncthreads();
    for (int off = 64; off > 0; off >>= 1) {
        if (t < off) red[t] += red[t + off];
        __syncthreads();
    }
    if (t == 0) rowss[nk] = red[0];
}

// -------------------------------------------------------------------------
// Kernel 5b: per-image global norm of the intra-normalized vector
// -------------------------------------------------------------------------
__global__ void k_gnorm(const float* __restrict__ rowss, float* __restrict__ gnorm) {
    __shared__ float red[64];
    const int n = blockIdx.x;
    const int t = threadIdx.x;   // 64 threads
    float ss = rowss[n * KCL + t];
    float rr = 1.f / fmaxf(sqrtf(ss), EPSN);
    red[t] = ss * rr * rr;
    __syncthreads();
    for (int off = 32; off > 0; off >>= 1) {
        if (t < off) red[t] += red[t + off];
        __syncthreads();
    }
    if (t == 0) gnorm[n] = 1.f / fmaxf(sqrtf(red[0]), EPSN);
}

// -------------------------------------------------------------------------
// Kernel 5c: fused intra-norm * global-norm scaling, in place on d_out
// -------------------------------------------------------------------------
__global__ void k_scale(float* __restrict__ out,
                        const float* __restrict__ rowss,
                        const float* __restrict__ gnorm) {
    size_t idx = (size_t)blockIdx.x * blockDim.x + threadIdx.x;
    if (idx >= (size_t)NB * DOUT) return;
    int n  = (int)(idx >> 15);
    int nk = (int)(idx >> 9);
    float rr = 1.f / fmaxf(sqrtf(rowss[nk]), EPSN);
    out[idx] = out[idx] * rr * gnorm[n];
}

// -------------------------------------------------------------------------
extern "C" void kernel_launch(void* const* d_in, const int* in_sizes, int n_in,
                              void* d_out, int out_size, void* d_ws, size_t ws_size,
                              hipStream_t stream) {
    const float* x         = (const float*)d_in[0];   // [32,512,30,40]
    const float* conv_w    = (const float*)d_in[1];   // [64,512]
    const float* centroids = (const float*)d_in[2];   // [64,512]
    float* out = (float*)d_out;                       // [32, 32768]

    // workspace layout (floats)
    float* ws     = (float*)d_ws;
    float* rnorm  = ws;                               // 38400
    float* a_buf  = rnorm + NB * HWP;                 // 2457600 (softmax * rnorm)
    float* ps     = a_buf + (size_t)NB * KCL * HWP;   // 32*25*64 = 51200
    float* asum   = ps + (size_t)NB * NT2 * KCL;      // 2048
    float* rowss  = asum + NB * KCL;                  // 2048
    float* gnorm  = rowss + NB * KCL;                 // 32

    k_rnorm<<<dim3((HWP + 255) / 256, NB), 256, 0, stream>>>(x, rnorm);
    k_logits_softmax<<<dim3(HWP / HTW, NB), 128, 0, stream>>>(x, conv_w, rnorm,
                                                              a_buf, ps);
    k_asum<<<NB, KCL, 0, stream>>>(ps, asum);
    k_vlad<<<dim3(CCH / 64, NB), 256, 0, stream>>>(x, a_buf, asum, centroids, out);
    k_rowss<<<NB * KCL, 128, 0, stream>>>(out, rowss);
    k_gnorm<<<NB, 64, 0, stream>>>(rowss, gnorm);
    k_scale<<<(NB * DOUT + 255) / 256, 256, 0, stream>>>(out, rowss, gnorm);
}